// RetrievalModel_44100724196047
// MI455X (gfx1250) — compile-verified
//
#include <hip/hip_runtime.h>
#include <hip/hip_bf16.h>

typedef float v2f __attribute__((ext_vector_type(2)));
typedef float v8f __attribute__((ext_vector_type(8)));

#define BATCH  2048
#define EMB    64
#define NCAND  100000
#define TOPK   100
#define NSLICE 4
#define SLICE  (NCAND / NSLICE)          // 25000
#define CHUNK  128
#define NCHUNK ((SLICE + CHUNK - 1) / CHUNK)   // 196

// ---------------------------------------------------------------------------
// CDNA5 async DMA helpers (ISA §15.18.3: GLOBAL_LOAD_ASYNC_TO_LDS_B128,
// tracked by ASYNCcnt; ISA §10.2: low 32 bits of a generic LDS pointer are the
// workgroup-relative LDS byte offset, which is exactly what VDST must hold).
// ---------------------------------------------------------------------------
__device__ __forceinline__ void async_load_b128(void* lds_ptr, const void* gptr) {
    unsigned lds_off = (unsigned)(size_t)lds_ptr;
    asm volatile("global_load_async_to_lds_b128 %0, %1, off"
                 :: "v"(lds_off), "v"(gptr) : "memory");
}
__device__ __forceinline__ void wait_asynccnt0() {
    asm volatile("s_wait_asynccnt 0x0" ::: "memory");
}

// ---------------------------------------------------------------------------
// Kernel 1: feature assembly (gathers + masked means)
// ---------------------------------------------------------------------------
__global__ void feat_kernel(const int* __restrict__ user_id,
                            const int* __restrict__ gender,
                            const int* __restrict__ age,
                            const int* __restrict__ cross_id,
                            const int* __restrict__ occ,
                            const int* __restrict__ movie_id,
                            const int* __restrict__ title_ids,
                            const int* __restrict__ genre_ids,
                            const float* __restrict__ user_t,
                            const float* __restrict__ gender_t,
                            const float* __restrict__ age_t,
                            const float* __restrict__ cross_t,
                            const float* __restrict__ occ_t,
                            const float* __restrict__ movie_t,
                            const float* __restrict__ title_t,
                            const float* __restrict__ genre_t,
                            float* __restrict__ qf,
                            float* __restrict__ cf)
{
    const int b = blockIdx.x;
    const int t = threadIdx.x;
    if (t < 320) {
        const int seg = t >> 6, d = t & 63;
        float v;
        if      (seg == 0) v = user_t  [user_id [b] * 64 + d];
        else if (seg == 1) v = gender_t[gender  [b] * 64 + d];
        else if (seg == 2) v = age_t   [age     [b] * 64 + d];
        else if (seg == 3) v = cross_t [cross_id[b] * 64 + d];
        else               v = occ_t   [occ     [b] * 64 + d];
        qf[b * 320 + t] = v;
    } else {
        const int c = t - 320;
        const int seg = c >> 6, d = c & 63;
        float v;
        if (seg == 0) {
            v = movie_t[movie_id[b] * 64 + d];
        } else if (seg == 1) {
            float s = 0.f; int cnt = 0;
            for (int k = 0; k < 16; ++k) {
                const int id = title_ids[b * 16 + k];
                if (id != 0) { s += title_t[id * 64 + d]; ++cnt; }
            }
            v = s / (float)(cnt > 0 ? cnt : 1);
        } else {
            float s = 0.f; int cnt = 0;
            for (int k = 0; k < 4; ++k) {
                const int id = genre_ids[b * 4 + k];
                if (id != 0) { s += genre_t[id * 64 + d]; ++cnt; }
            }
            v = s / (float)(cnt > 0 ? cnt : 1);
        }
        cf[b * 192 + c] = v;
    }
}

// ---------------------------------------------------------------------------
// Kernel 2: C[M,N] = act(A[M,K] @ W[K,N] + bias[N]) via V_WMMA_F32_16X16X4_F32
// ---------------------------------------------------------------------------
__global__ void gemm_wmma(const float* __restrict__ A,
                          const float* __restrict__ W,
                          const float* __restrict__ bias,
                          float* __restrict__ C,
                          int M, int K, int N, int relu)
{
    const int lane  = threadIdx.x & 31;
    const int wave  = threadIdx.x >> 5;
    const int ntile = blockIdx.x * 4 + wave;
    const int mtile = blockIdx.y;
    if (ntile * 16 >= N) return;            // wave-uniform: EXEC stays all-1s

    const int half = lane >> 4;
    const int lm   = lane & 15;
    const int row  = mtile * 16 + lm;
    const int col  = ntile * 16 + lm;
    const int koff = half * 2;

    v8f acc = {};
    for (int kb = 0; kb < K; kb += 4) {
        v2f a, bfr;
        a.x   = A[row * K + kb + koff];
        a.y   = A[row * K + kb + koff + 1];
        bfr.x = W[(kb + koff)     * N + col];
        bfr.y = W[(kb + koff + 1) * N + col];
        acc = __builtin_amdgcn_wmma_f32_16x16x4_f32(
                  false, a, false, bfr, (short)0, acc, false, false);
    }

    const float bv = bias[col];
#pragma unroll
    for (int j = 0; j < 8; ++j) {
        const int r = mtile * 16 + j + half * 8;
        float v = acc[j] + bv;
        if (relu) v = fmaxf(v, 0.0f);
        C[r * N + col] = v;
    }
}

// ---------------------------------------------------------------------------
// Kernel 3: scoring + per-slice top-100.
// Grid: (BATCH/16, NSLICE). Block = 256 threads (8 waves), 16 query rows.
// Per 128-candidate chunk:
//   - candidate rows DMA'd into LDS via global_load_async_to_lds_b128
//     (coalesced 16B/lane; ASYNCcnt + barrier for visibility)
//   - 8 waves compute a 16x128 score tile with 16 chained f32 WMMAs
//   - next chunk's DMA issued BEFORE selection -> DMA hides under the
//     serial heap phase (single-buffer software pipeline)
//   - 16 threads stream the tile into per-row size-100 LDS min-heaps
// Static LDS ~53 KB (cbuf 32K + sc 8K + heaps ~12.8K).
// ---------------------------------------------------------------------------
__global__ void score_topk(const float* __restrict__ Q,      // [BATCH,64]
                           const float* __restrict__ cand,   // [NCAND,64]
                           float* __restrict__ wval,         // [BATCH, NSLICE*TOPK]
                           int*   __restrict__ widx)         // [BATCH, NSLICE*TOPK]
{
    __shared__ float cbuf[CHUNK * 64];      // 32 KB candidate chunk
    __shared__ float sc[16 * CHUNK];        // 8 KB score tile
    __shared__ float hval[16 * TOPK];
    __shared__ int   hidx[16 * TOPK];
    __shared__ int   hcnt[16];

    const int tid  = threadIdx.x;
    const int lane = tid & 31;
    const int wave = tid >> 5;
    const int half = lane >> 4;
    const int lm   = lane & 15;
    const int rowbase = blockIdx.x * 16;
    const int slice   = blockIdx.y;
    const int n0      = slice * SLICE;

    if (tid < 16) hcnt[tid] = 0;

    // Preload A fragments for this block's 16 query rows (K=64 -> 16 wmma steps)
    v2f afr[16];
    const int arow = rowbase + lm;
#pragma unroll
    for (int s = 0; s < 16; ++s) {
        afr[s].x = Q[arow * 64 + 4 * s + 2 * half];
        afr[s].y = Q[arow * 64 + 4 * s + 2 * half + 1];
    }

    // Issue async DMA for a chunk: each wave moves its 16 rows (4 KB) in
    // 8 x b128 coalesced transfers.
    auto issue = [&](int ch) {
        const int lbase = ch * CHUNK + wave * 16;   // slice-local col base
#pragma unroll
        for (int it = 0; it < 8; ++it) {
            const int li  = it * 32 + lane;         // 0..255
            const int r   = li >> 4;                // row within wave's 16
            const int seg = li & 15;                // 16B segment within row
            int gr = n0 + lbase + r;
            if (gr > NCAND - 1) gr = NCAND - 1;
            async_load_b128(&cbuf[(wave * 16 + r) * 64 + seg * 4],
                            &cand[(size_t)gr * 64 + seg * 4]);
        }
    };

    issue(0);
    for (int ch = 0; ch < NCHUNK; ++ch) {
        wait_asynccnt0();                   // this wave's DMA done
        __syncthreads();                    // all waves' DMA visible; sc free

        // 16x16 tile per wave from LDS-resident candidates
        const int lcol = ch * CHUNK + wave * 16 + lm;   // slice-local column
        const float* cb = &cbuf[(wave * 16 + lm) * 64];
        v8f acc = {};
#pragma unroll
        for (int s = 0; s < 16; ++s) {
            v2f bfr;
            bfr.x = cb[4 * s + 2 * half];
            bfr.y = cb[4 * s + 2 * half + 1];
            acc = __builtin_amdgcn_wmma_f32_16x16x4_f32(
                      false, afr[s], false, bfr, (short)0, acc, false, false);
        }
#pragma unroll
        for (int j = 0; j < 8; ++j) {
            const int r = j + half * 8;
            sc[r * CHUNK + wave * 16 + lm] = (lcol < SLICE) ? acc[j] : -3.0e38f;
        }
        __syncthreads();                    // sc ready; cbuf fully consumed

        if (ch + 1 < NCHUNK) issue(ch + 1); // DMA overlaps heap selection

        if (tid < 16) {
            const int r = tid;
            float* hv = &hval[r * TOPK];
            int*   hi = &hidx[r * TOPK];
            int cnt = hcnt[r];
            for (int i = 0; i < CHUNK; ++i) {
                const float v  = sc[r * CHUNK + i];
                const int   id = n0 + ch * CHUNK + i;
                if (cnt < TOPK) {
                    int p = cnt++;
                    hv[p] = v; hi[p] = id;
                    while (p > 0) {                       // sift up (min-heap)
                        const int par = (p - 1) >> 1;
                        if (hv[par] > hv[p]) {
                            float tv = hv[par]; hv[par] = hv[p]; hv[p] = tv;
                            int   ti = hi[par]; hi[par] = hi[p]; hi[p] = ti;
                            p = par;
                        } else break;
                    }
                } else if (v > hv[0]) {
                    hv[0] = v; hi[0] = id;
                    int p = 0;
                    for (;;) {                            // sift down
                        const int l = 2 * p + 1, rr = 2 * p + 2;
                        int m = p;
                        if (l  < TOPK && hv[l]  < hv[m]) m = l;
                        if (rr < TOPK && hv[rr] < hv[m]) m = rr;
                        if (m == p) break;
                        float tv = hv[m]; hv[m] = hv[p]; hv[p] = tv;
                        int   ti = hi[m]; hi[m] = hi[p]; hi[p] = ti;
                        p = m;
                    }
                }
            }
            hcnt[r] = cnt;
        }
        // next iteration's wait + barrier separates selection from sc rewrite
    }

    // Dump per-slice top-100 (unsorted; merge kernel re-selects + sorts)
    if (tid < 16) {
        const int r = tid;
        const int cnt = hcnt[r];
        float* hv = &hval[r * TOPK];
        int*   hi = &hidx[r * TOPK];
        const size_t base = (size_t)(rowbase + r) * (NSLICE * TOPK) + slice * TOPK;
        for (int k = 0; k < TOPK; ++k) {
            const int kk = (k < cnt) ? k : 0;
            wval[base + k] = hv[kk];
            widx[base + k] = hi[kk];
        }
    }
}

// ---------------------------------------------------------------------------
// Kernel 4: merge NSLICE partial top-100 lists per row -> final sorted top-100.
// 64 threads/block, one row per thread, LDS heaps (51.2 KB).
// ---------------------------------------------------------------------------
__global__ void merge_topk(const float* __restrict__ wval,
                           const int*   __restrict__ widx,
                           float* __restrict__ pred)
{
    __shared__ float mhv[64 * TOPK];
    __shared__ int   mhi[64 * TOPK];
    const int t   = threadIdx.x;
    const int row = blockIdx.x * 64 + t;
    if (row >= BATCH) return;

    float* hv = &mhv[t * TOPK];
    int*   hi = &mhi[t * TOPK];
    const size_t base = (size_t)row * (NSLICE * TOPK);
    int cnt = 0;
    for (int i = 0; i < NSLICE * TOPK; ++i) {
        const float v  = wval[base + i];
        const int   id = widx[base + i];
        if (cnt < TOPK) {
            int p = cnt++;
            hv[p] = v; hi[p] = id;
            while (p > 0) {
                const int par = (p - 1) >> 1;
                if (hv[par] > hv[p]) {
                    float tv = hv[par]; hv[par] = hv[p]; hv[p] = tv;
                    int   ti = hi[par]; hi[par] = hi[p]; hi[p] = ti;
                    p = par;
                } else break;
            }
        } else if (v > hv[0]) {
            hv[0] = v; hi[0] = id;
            int p = 0;
            for (;;) {
                const int l = 2 * p + 1, rr = 2 * p + 2;
                int m = p;
                if (l  < TOPK && hv[l]  < hv[m]) m = l;
                if (rr < TOPK && hv[rr] < hv[m]) m = rr;
                if (m == p) break;
                float tv = hv[m]; hv[m] = hv[p]; hv[p] = tv;
                int   ti = hi[m]; hi[m] = hi[p]; hi[p] = ti;
                p = m;
            }
        }
    }
    // in-place min-heap heapsort -> descending order from index 0
    for (int sz = cnt; sz > 1; ) {
        --sz;
        float tv = hv[0]; hv[0] = hv[sz]; hv[sz] = tv;
        int   ti = hi[0]; hi[0] = hi[sz]; hi[sz] = ti;
        int p = 0;
        for (;;) {
            const int l = 2 * p + 1, rr = 2 * p + 2;
            int m = p;
            if (l  < sz && hv[l]  < hv[m]) m = l;
            if (rr < sz && hv[rr] < hv[m]) m = rr;
            if (m == p) break;
            tv = hv[m]; hv[m] = hv[p]; hv[p] = tv;
            ti = hi[m]; hi[m] = hi[p]; hi[p] = ti;
            p = m;
        }
    }
    for (int k = 0; k < TOPK; ++k)
        pred[(size_t)row * TOPK + k] = (float)hi[k];
}

// ---------------------------------------------------------------------------
extern "C" void kernel_launch(void* const* d_in, const int* in_sizes, int n_in,
                              void* d_out, int out_size, void* d_ws, size_t ws_size,
                              hipStream_t stream)
{
    // setup_inputs() dict order (interleaved qWi,qbi,cWi,cbi!)
    const int*   user_id   = (const int*)  d_in[0];
    const int*   gender    = (const int*)  d_in[1];
    const int*   age       = (const int*)  d_in[2];
    const int*   cross_id  = (const int*)  d_in[3];
    const int*   occ       = (const int*)  d_in[4];
    const int*   movie_id  = (const int*)  d_in[5];
    const int*   title_ids = (const int*)  d_in[6];
    const int*   genre_ids = (const int*)  d_in[7];
    const float* user_t    = (const float*)d_in[8];
    const float* gender_t  = (const float*)d_in[9];
    const float* age_t     = (const float*)d_in[10];
    const float* cross_t   = (const float*)d_in[11];
    const float* occ_t     = (const float*)d_in[12];
    const float* movie_t   = (const float*)d_in[13];
    const float* title_t   = (const float*)d_in[14];
    const float* genre_t   = (const float*)d_in[15];
    const float* qW0 = (const float*)d_in[16];
    const float* qb0 = (const float*)d_in[17];
    const float* cW0 = (const float*)d_in[18];
    const float* cb0 = (const float*)d_in[19];
    const float* qW1 = (const float*)d_in[20];
    const float* qb1 = (const float*)d_in[21];
    const float* cW1 = (const float*)d_in[22];
    const float* cb1 = (const float*)d_in[23];
    const float* qW2 = (const float*)d_in[24];
    const float* qb2 = (const float*)d_in[25];
    const float* cW2 = (const float*)d_in[26];
    const float* cb2 = (const float*)d_in[27];
    const float* cand_table = (const float*)d_in[28];

    // Workspace carve-out (~14 MB)
    float* qf   = (float*)d_ws;                     // [2048, 320]
    float* cf   = qf   + (size_t)BATCH * 320;       // [2048, 192]
    float* h0   = cf   + (size_t)BATCH * 192;       // [2048, 256]
    float* h1   = h0   + (size_t)BATCH * 256;       // [2048, 128]
    float* wval = h1   + (size_t)BATCH * 128;       // [2048, 400]
    int*   widx = (int*)(wval + (size_t)BATCH * NSLICE * TOPK); // [2048, 400]

    // Output layout: query_emb | cand_emb | predictions(ids as float)
    float* qemb = (float*)d_out;
    float* cemb = qemb + (size_t)BATCH * EMB;
    float* pred = cemb + (size_t)BATCH * EMB;

    feat_kernel<<<BATCH, 512, 0, stream>>>(user_id, gender, age, cross_id, occ,
                                           movie_id, title_ids, genre_ids,
                                           user_t, gender_t, age_t, cross_t, occ_t,
                                           movie_t, title_t, genre_t, qf, cf);

    const dim3 blk(128);
    // Query tower: 320 -> 256 -> 128 -> 64
    gemm_wmma<<<dim3(256 / 64, BATCH / 16), blk, 0, stream>>>(qf, qW0, qb0, h0,  BATCH, 320, 256, 1);
    gemm_wmma<<<dim3(128 / 64, BATCH / 16), blk, 0, stream>>>(h0, qW1, qb1, h1,  BATCH, 256, 128, 1);
    gemm_wmma<<<dim3( 64 / 64, BATCH / 16), blk, 0, stream>>>(h1, qW2, qb2, qemb, BATCH, 128,  64, 0);
    // Candidate tower: 192 -> 256 -> 128 -> 64
    gemm_wmma<<<dim3(256 / 64, BATCH / 16), blk, 0, stream>>>(cf, cW0, cb0, h0,  BATCH, 192, 256, 1);
    gemm_wmma<<<dim3(128 / 64, BATCH / 16), blk, 0, stream>>>(h0, cW1, cb1, h1,  BATCH, 256, 128, 1);
    gemm_wmma<<<dim3( 64 / 64, BATCH / 16), blk, 0, stream>>>(h1, cW2, cb2, cemb, BATCH, 128,  64, 0);

    // Sliced brute-force scoring + per-slice top-100 (512 workgroups)
    score_topk<<<dim3(BATCH / 16, NSLICE), 256, 0, stream>>>(qemb, cand_table, wval, widx);
    // Merge slices -> final sorted top-100 ids
    merge_topk<<<dim3(BATCH / 64), 64, 0, stream>>>(wval, widx, pred);
}